// LSTMSequenceModel_40235253629450
// MI455X (gfx1250) — compile-verified
//
#include <hip/hip_runtime.h>
#include <math.h>

// Problem constants (fixed by the reference).
constexpr int N_NODES = 50000;
constexpr int F_IN    = 128;
constexpr int HD      = 256;
constexpr int K_CHEB  = 3;
constexpr int T_STEPS = 7;
constexpr int KX      = K_CHEB * F_IN;          // 384 rows from theta_x
constexpr int KH      = K_CHEB * HD;            // 768 rows from theta_h
constexpr int KTOT    = KX + KH;                // 1152

typedef __attribute__((ext_vector_type(2))) float v2f;
typedef __attribute__((ext_vector_type(4))) float v4f;
typedef __attribute__((ext_vector_type(8))) float v8f;

__device__ __forceinline__ float disv(float d) {
  return d > 0.f ? rsqrtf(fmaxf(d, 1e-20f)) : 0.f;
}
__device__ __forceinline__ float sigm(float x) { return 1.f / (1.f + expf(-x)); }

// ---------------------------------------------------------------- zero scratch
__global__ void k_zero(float* __restrict__ p, int n) {
  int i = blockIdx.x * blockDim.x + threadIdx.x;
  if (i < n) p[i] = 0.f;
}

// ---------------------------------------------------------------- deg[src] += w (skip self loops)
__global__ void k_deg(const int* __restrict__ ei, const float* __restrict__ ew,
                      float* __restrict__ deg, int E) {
  int e = blockIdx.x * blockDim.x + threadIdx.x;
  if (e >= E) return;
  int s = ei[e], d = ei[E + e];
  if (s != d) atomicAdd(&deg[s], ew[e]);
}

// ---------------------------------------------------------------- edges into node 0:
// cc[src] += w_norm ; acc1_* += w_norm * feat[src]   (Tx1[0] = -acc1)
__global__ void k_c_acc1(const int* __restrict__ ei, const float* __restrict__ ew,
                         const float* __restrict__ deg,
                         const float* __restrict__ x, const float* __restrict__ h,
                         float* __restrict__ cc, float* __restrict__ acc1x,
                         float* __restrict__ acc1h, int E) {
  int e = blockIdx.x * blockDim.x + threadIdx.x;
  if (e >= E) return;
  int s = ei[e], d = ei[E + e];
  if (d != 0 || s == d) return;                     // only non-self edges into node 0
  float wn = disv(deg[s]) * ew[e] * disv(deg[0]);
  atomicAdd(&cc[s], wn);
  const float* xs = x + (size_t)s * F_IN;
  for (int f = 0; f < F_IN; ++f) atomicAdd(&acc1x[f], wn * xs[f]);
  const float* hs = h + (size_t)s * HD;
  for (int j = 0; j < HD; ++j) atomicAdd(&acc1h[j], wn * hs[j]);
}

// ---------------------------------------------------------------- edges into neighbors of 0:
// acc2_* += cc[dst] * w_norm * feat[src]   (Tx2[0] = 2*acc2 - Tx0[0])
__global__ void k_acc2(const int* __restrict__ ei, const float* __restrict__ ew,
                       const float* __restrict__ deg, const float* __restrict__ cc,
                       const float* __restrict__ x, const float* __restrict__ h,
                       float* __restrict__ acc2x, float* __restrict__ acc2h, int E) {
  int e = blockIdx.x * blockDim.x + threadIdx.x;
  if (e >= E) return;
  int s = ei[e], d = ei[E + e];
  if (s == d) return;
  float cd = cc[d];
  if (cd == 0.f) return;                            // exact: non-neighbors contribute 0
  float coef = cd * disv(deg[s]) * ew[e] * disv(deg[d]);
  const float* xs = x + (size_t)s * F_IN;
  for (int f = 0; f < F_IN; ++f) atomicAdd(&acc2x[f], coef * xs[f]);
  const float* hs = h + (size_t)s * HD;
  for (int j = 0; j < HD; ++j) atomicAdd(&acc2h[j], coef * hs[j]);
}

// ---------------------------------------------------------------- node-0 gates via fp32 WMMA
// 64 tiles (4 gates x 16 col-tiles), one wave32 per tile.
// Two straight-line K-loops (theta_x then theta_h): per iteration 1x ds_load_b128,
// 2x global_load_b32 (pointer-increment addressing), 4 FMAs, 1 v_wmma_f32_16x16x4_f32.
__global__ void k_gates(const float* __restrict__ x, const float* __restrict__ h,
                        const float* __restrict__ thx, const float* __restrict__ thh,
                        const float* __restrict__ bx, const float* __restrict__ bh,
                        const float* __restrict__ acc1x, const float* __restrict__ acc2x,
                        const float* __restrict__ acc1h, const float* __restrict__ acc2h,
                        float* __restrict__ gbuf) {
  __shared__ __align__(16) float u[KTOT];
  for (int i = threadIdx.x; i < KTOT; i += blockDim.x) {
    float v;
    if (i < F_IN)            v = x[i];                                         // Tx0
    else if (i < 2 * F_IN)   v = -acc1x[i - F_IN];                             // Tx1
    else if (i < 3 * F_IN)   v = 2.f * acc2x[i - 2 * F_IN] - x[i - 2 * F_IN];  // Tx2
    else {
      int j = i - KX;
      if (j < HD)            v = h[j];
      else if (j < 2 * HD)   v = -acc1h[j - HD];
      else                   v = 2.f * acc2h[j - 2 * HD] - h[j - 2 * HD];
    }
    u[i] = v;
  }
  __syncthreads();

  const int lane = threadIdx.x & 31;
  const int wave = threadIdx.x >> 5;
  const int tile = blockIdx.x * (blockDim.x >> 5) + wave;  // 0..63
  const int gate = tile >> 4;
  const int n0   = (tile & 15) << 4;
  const int half = lane >> 4;                              // 0 | 1
  const int col  = n0 + (lane & 15);

  // Branch-free A-fragment masks: only lane 0 (K0/K1) and lane 16 (K2/K3) carry row M=0.
  const float m0  = (lane == 0)  ? 1.f : 0.f;
  const float m16 = (lane == 16) ? 1.f : 0.f;

  // B pointers: row (k0 + 2*half) at bp[0], row (k0 + 2*half + 1) at bp[HD].
  const float* bpx = thx + ((size_t)(gate * KX) + (size_t)(half << 1)) * HD + col;
  const float* bph = thh + ((size_t)(gate * KH) + (size_t)(half << 1)) * HD + col;

  v8f acc = {0.f, 0.f, 0.f, 0.f, 0.f, 0.f, 0.f, 0.f};

#pragma unroll 4
  for (int k0 = 0; k0 < KX; k0 += 4) {
    v4f uv = *(const v4f*)&u[k0];                          // one ds_load_b128 (broadcast)
    v2f a;
    a.x = m0 * uv.x + m16 * uv.z;
    a.y = m0 * uv.y + m16 * uv.w;
    v2f b;
    b.x = bpx[0];
    b.y = bpx[HD];
    acc = __builtin_amdgcn_wmma_f32_16x16x4_f32(false, a, false, b, (short)0, acc,
                                                false, false);
    bpx += 4 * HD;
  }
#pragma unroll 4
  for (int k0 = 0; k0 < KH; k0 += 4) {
    v4f uv = *(const v4f*)&u[KX + k0];
    v2f a;
    a.x = m0 * uv.x + m16 * uv.z;
    a.y = m0 * uv.y + m16 * uv.w;
    v2f b;
    b.x = bph[0];
    b.y = bph[HD];
    acc = __builtin_amdgcn_wmma_f32_16x16x4_f32(false, a, false, b, (short)0, acc,
                                                false, false);
    bph += 4 * HD;
  }

  // D layout: lanes 0-15 hold N=lane, VGPR0 = row M=0.
  if (lane < 16) {
    int o = gate * HD + col;
    gbuf[o] = acc[0] + bx[o] + bh[o];
  }
}

// ---------------------------------------------------------------- GConvLSTM cell @ node 0
__global__ void k_cell(const float* __restrict__ gbuf, const float* __restrict__ c_in,
                       const float* __restrict__ w_c, const float* __restrict__ b_gate,
                       float* __restrict__ hn0, float* __restrict__ cn0) {
  int j = threadIdx.x;
  if (j >= HD) return;
  float cv = c_in[j];   // row 0 of c
  float g0 = gbuf[j], g1 = gbuf[HD + j], g2 = gbuf[2 * HD + j], g3 = gbuf[3 * HD + j];
  float Ig = sigm(g0 + w_c[j] * cv + b_gate[j]);
  float Fg = sigm(g1 + w_c[HD + j] * cv + b_gate[HD + j]);
  float Cn = Fg * cv + Ig * tanhf(g2 + b_gate[2 * HD + j]);
  float Og = sigm(g3 + w_c[2 * HD + j] * Cn + b_gate[3 * HD + j]);
  hn0[j] = Og * tanhf(Cn);
  cn0[j] = Cn;
}

// ---------------------------------------------------------------- T=7 sequential scalar-input LSTM
__global__ void k_lstm(const float* __restrict__ x,
                       const float* __restrict__ hn0, const float* __restrict__ cn0,
                       const float* __restrict__ w_ih, const float* __restrict__ w_hh,
                       const float* __restrict__ b_ih, const float* __restrict__ b_hh,
                       const float* __restrict__ w_out, const float* __restrict__ b_out,
                       float* __restrict__ out) {
  __shared__ float hs[HD];
  __shared__ float red;
  __shared__ float inp_s;
  int j = threadIdx.x;
  float cj = cn0[j];
  hs[j] = hn0[j];
  if (j == 0) inp_s = x[F_IN - 1];   // x[0, -1]
  __syncthreads();
  float wo = w_out[j];
  for (int t = 0; t < T_STEPS; ++t) {
    float inp = inp_s;
    float gv[4];
#pragma unroll
    for (int gi = 0; gi < 4; ++gi) {
      int row = gi * HD + j;
      const float* wr = w_hh + (size_t)row * HD;
      float a = b_ih[row] + b_hh[row] + w_ih[row] * inp;
      for (int hh = 0; hh < HD; ++hh) a += wr[hh] * hs[hh];
      gv[gi] = a;
    }
    float Ig = sigm(gv[0]), Fg = sigm(gv[1]), Gg = tanhf(gv[2]), Og = sigm(gv[3]);
    cj = Fg * cj + Ig * Gg;
    float hj = Og * tanhf(cj);
    __syncthreads();
    hs[j] = hj;
    if (j == 0) red = 0.f;
    __syncthreads();
    atomicAdd(&red, wo * hj);
    __syncthreads();
    if (j == 0) {
      float o = red + b_out[0];
      out[t] = o;
      inp_s = o;                     // next step's input is this step's output
    }
    __syncthreads();
  }
}

// ---------------------------------------------------------------- launch
extern "C" void kernel_launch(void* const* d_in, const int* in_sizes, int n_in,
                              void* d_out, int out_size, void* d_ws, size_t ws_size,
                              hipStream_t stream) {
  (void)n_in; (void)out_size; (void)ws_size;
  const float* x      = (const float*)d_in[0];
  const int*   ei     = (const int*)d_in[1];
  const float* ew     = (const float*)d_in[2];
  const float* h      = (const float*)d_in[3];
  const float* c      = (const float*)d_in[4];
  const float* thx    = (const float*)d_in[5];
  const float* bx     = (const float*)d_in[6];
  const float* thh    = (const float*)d_in[7];
  const float* bh     = (const float*)d_in[8];
  const float* w_c    = (const float*)d_in[9];
  const float* b_gate = (const float*)d_in[10];
  const float* w_ih   = (const float*)d_in[11];
  const float* w_hh   = (const float*)d_in[12];
  const float* b_ih   = (const float*)d_in[13];
  const float* b_hh   = (const float*)d_in[14];
  const float* w_out  = (const float*)d_in[15];
  const float* b_out  = (const float*)d_in[16];
  float* out = (float*)d_out;
  const int E = in_sizes[2];

  float* wsf   = (float*)d_ws;
  float* deg   = wsf;                   // N
  float* cc    = wsf + N_NODES;         // N
  float* acc1x = wsf + 2 * N_NODES;     // 128
  float* acc2x = acc1x + F_IN;          // 128
  float* acc1h = acc2x + F_IN;          // 256
  float* acc2h = acc1h + HD;            // 256
  float* gbuf  = acc2h + HD;            // 1024
  float* hn0   = gbuf + 4 * HD;         // 256
  float* cn0   = hn0 + HD;              // 256

  const int nz = 2 * N_NODES + 2 * F_IN + 2 * HD;
  k_zero<<<(nz + 255) / 256, 256, 0, stream>>>(wsf, nz);
  const int eb = (E + 255) / 256;
  k_deg<<<eb, 256, 0, stream>>>(ei, ew, deg, E);
  k_c_acc1<<<eb, 256, 0, stream>>>(ei, ew, deg, x, h, cc, acc1x, acc1h, E);
  k_acc2<<<eb, 256, 0, stream>>>(ei, ew, deg, cc, x, h, acc2x, acc2h, E);
  k_gates<<<8, 256, 0, stream>>>(x, h, thx, thh, bx, bh,
                                 acc1x, acc2x, acc1h, acc2h, gbuf);
  k_cell<<<1, HD, 0, stream>>>(gbuf, c, w_c, b_gate, hn0, cn0);
  k_lstm<<<1, HD, 0, stream>>>(x, hn0, cn0, w_ih, w_hh, b_ih, b_hh, w_out, b_out, out);
}